// LiPSRDFMAE_86388972192375
// MI455X (gfx1250) — compile-verified
//
#include <hip/hip_runtime.h>
#include <math.h>

typedef float v2f __attribute__((ext_vector_type(2)));
typedef float v8f __attribute__((ext_vector_type(8)));

#define T_FRAMES 100
#define R_REP    8
#define N_ATOMS  83
#define NPAD     96      // padded to 6 tiles of 16
#define NTILE    6
#define NSHIFT   27
#define NBINS    120

#if __has_builtin(__builtin_amdgcn_global_load_async_to_lds_b32)
#define HAVE_ASYNC_LDS 1
#else
#define HAVE_ASYNC_LDS 0
#endif

__device__ __forceinline__ void wait_async_zero() {
#if __has_builtin(__builtin_amdgcn_s_wait_asynccnt)
    __builtin_amdgcn_s_wait_asynccnt(0);
#else
    asm volatile("s_wait_asynccnt 0x0" ::: "memory");
#endif
}

// One block per (frame t, replica r). 256 threads = 8 waves.
// Gram matrix G = X X^T via V_WMMA_F32_16X16X4_F32 (K=3 padded to 4);
// min-image handled with precomputed per-shift LDS tables.
__global__ __launch_bounds__(256)
void rdf_hist_kernel(const float* __restrict__ pos,    // [T][R][N][3]
                     const float* __restrict__ cell,   // [3][3] row-major
                     const float* __restrict__ bins,   // [NBINS+1]
                     unsigned int* __restrict__ counts) // [R][NBINS]
{
    __shared__ float Xs[NPAD * 3];
    __shared__ float nrm[NPAD];
    __shared__ float eS[NSHIFT * NPAD];   // 2 * x_j . s
    __shared__ float cS[NSHIFT * NPAD];   // |s|^2 - 2 * x_i . s
    __shared__ float sh[NSHIFT][3];
    __shared__ float s2[NSHIFT];
    __shared__ unsigned int hist[NBINS];

    const int tid = threadIdx.x;
    const int t = blockIdx.x / R_REP;
    const int r = blockIdx.x % R_REP;
    const float* x = pos + ((size_t)(t * R_REP + r)) * (N_ATOMS * 3);

#if HAVE_ASYNC_LDS
    // CDNA5 async path: stream positions straight into LDS (ASYNCcnt-tracked),
    // no VGPR round-trip. One dword per lane; tid 0..248 covers the tile.
    if (tid < N_ATOMS * 3) {
        auto gsrc_v = (const __attribute__((address_space(1))) void*)(x + tid);
        auto ldst_v = (__attribute__((address_space(3))) void*)(&Xs[tid]);
        auto gsrc = (__attribute__((address_space(1))) int*)gsrc_v;
        auto ldst = (__attribute__((address_space(3))) int*)ldst_v;
        __builtin_amdgcn_global_load_async_to_lds_b32(gsrc, ldst, 0, 0);
    }
    for (int k = N_ATOMS * 3 + tid; k < NPAD * 3; k += 256)
        Xs[k] = 0.0f;
    wait_async_zero();
#else
    for (int i = tid; i < NPAD * 3; i += 256)
        Xs[i] = (i < N_ATOMS * 3) ? x[i] : 0.0f;
#endif

    // 27 periodic shifts: s = a*cell[0] + b*cell[1] + c*cell[2], a,b,c in {-1,0,1}
    if (tid < NSHIFT) {
        const int a = tid % 3 - 1;
        const int b = (tid / 3) % 3 - 1;
        const int c = tid / 9 - 1;
        const float sx = a * cell[0] + b * cell[3] + c * cell[6];
        const float sy = a * cell[1] + b * cell[4] + c * cell[7];
        const float sz = a * cell[2] + b * cell[5] + c * cell[8];
        sh[tid][0] = sx; sh[tid][1] = sy; sh[tid][2] = sz;
        s2[tid] = sx * sx + sy * sy + sz * sz;
    }
    if (tid < NBINS) hist[tid] = 0u;
    __syncthreads();

    for (int k = tid; k < NSHIFT * NPAD; k += 256) {
        const int s = k / NPAD, i = k % NPAD;
        const float px = Xs[i * 3], py = Xs[i * 3 + 1], pz = Xs[i * 3 + 2];
        const float e = 2.0f * (px * sh[s][0] + py * sh[s][1] + pz * sh[s][2]);
        eS[k] = e;
        cS[k] = s2[s] - e;
    }
    for (int i = tid; i < NPAD; i += 256) {
        const float px = Xs[i * 3], py = Xs[i * 3 + 1], pz = Xs[i * 3 + 2];
        nrm[i] = px * px + py * py + pz * pz;
    }
    __syncthreads();

    const float b0    = bins[0];
    const float blast = bins[NBINS];
    const float invw  = (float)NBINS / (blast - b0);

    const int wave = tid >> 5;
    const int lane = tid & 31;
    const int lrow = lane & 15;
    const int lhi  = lane >> 4;
    // Branchless fragment addressing: lanes 0-15 read K=0 (and K=1),
    // lanes 16-31 read K=2 (and K=3 == 0). Avoids EXEC churn before WMMA.
    const int colA = lhi ? 2 : 0;

    for (int tile = wave; tile < NTILE * NTILE; tile += 8) {
        const int ti = tile / NTILE, tj = tile % NTILE;
        const int i0 = ti * 16, j0 = tj * 16;

        const int ra = (i0 + lrow) * 3;
        const int rb = (j0 + lrow) * 3;
        v2f A, B;
        A.x = Xs[ra + colA];
        A.y = lhi ? 0.0f : Xs[ra + 1];
        B.x = Xs[rb + colA];
        B.y = lhi ? 0.0f : Xs[rb + 1];

        v8f Cz = {};
        v8f G = __builtin_amdgcn_wmma_f32_16x16x4_f32(false, A, false, B,
                                                      (short)0, Cz, false, false);

        const int j = j0 + lrow;            // column index for this lane (fixed)
        float ej[NSHIFT];
        #pragma unroll
        for (int s = 0; s < NSHIFT; ++s) ej[s] = eS[s * NPAD + j];
        const float nj = nrm[j];

        if (j < N_ATOMS) {
            #pragma unroll
            for (int v = 0; v < 8; ++v) {
                const int i = i0 + v + (lhi ? 8 : 0);
                if (i < N_ATOMS && i != j) {
                    const float base = nrm[i] + nj - 2.0f * G[v];
                    float mn = cS[0 * NPAD + i] + ej[0];
                    #pragma unroll
                    for (int s = 1; s < NSHIFT; ++s)
                        mn = fminf(mn, cS[s * NPAD + i] + ej[s]);
                    const float d2 = base + mn;
                    const float d = sqrtf(fmaxf(d2, 0.0f));
                    if (d >= b0 && d <= blast) {
                        int idx = (int)((d - b0) * invw);
                        idx = idx < (NBINS - 1) ? idx : (NBINS - 1);
                        atomicAdd(&hist[idx], 1u);
                    }
                }
            }
        }
    }
    __syncthreads();
    if (tid < NBINS)
        atomicAdd(&counts[r * NBINS + tid], hist[tid]);
}

// One block per replica: rdf = counts / Z, mae = XLIM * mean|rdf - gt|.
__global__ __launch_bounds__(128)
void rdf_finalize_kernel(const float* __restrict__ cell,
                         const float* __restrict__ gt,
                         const float* __restrict__ bins,
                         const unsigned int* __restrict__ counts,
                         float* __restrict__ out)
{
    __shared__ float red[128];
    const int r = blockIdx.x;
    const int tid = threadIdx.x;

    const float c00 = cell[0], c01 = cell[1], c02 = cell[2];
    const float c10 = cell[3], c11 = cell[4], c12 = cell[5];
    const float c20 = cell[6], c21 = cell[7], c22 = cell[8];
    const float cx = c01 * c12 - c02 * c11;
    const float cy = c02 * c10 - c00 * c12;
    const float cz = c00 * c11 - c01 * c10;
    const float vol = fabsf(cx * c20 + cy * c21 + cz * c22);
    const float rho = (float)(T_FRAMES * N_ATOMS * N_ATOMS) / vol;
    const float coef = rho * (4.0f / 3.0f) * 3.14159265358979323846f;

    float a = 0.0f;
    if (tid < NBINS) {
        const float lo = bins[tid], hi = bins[tid + 1];
        const float Z = coef * (hi * hi * hi - lo * lo * lo);
        const float rdf = (float)counts[r * NBINS + tid] / Z;
        out[r * NBINS + tid] = rdf;
        a = fabsf(rdf - gt[tid]);
    }
    red[tid] = a;
    __syncthreads();
    for (int s = 64; s > 0; s >>= 1) {
        if (tid < s) red[tid] += red[tid + s];
        __syncthreads();
    }
    if (tid == 0)
        out[R_REP * NBINS + r] = 6.0f * red[0] / (float)NBINS;
}

extern "C" void kernel_launch(void* const* d_in, const int* in_sizes, int n_in,
                              void* d_out, int out_size, void* d_ws, size_t ws_size,
                              hipStream_t stream) {
    (void)in_sizes; (void)n_in; (void)out_size; (void)ws_size;
    const float* pos  = (const float*)d_in[0];   // [100,8,83,3]
    const float* cell = (const float*)d_in[1];   // [3,3]
    const float* gt   = (const float*)d_in[2];   // [120]
    const float* bins = (const float*)d_in[3];   // [121]
    unsigned int* counts = (unsigned int*)d_ws;  // [8][120] u32

    (void)hipMemsetAsync(counts, 0, R_REP * NBINS * sizeof(unsigned int), stream);
    rdf_hist_kernel<<<T_FRAMES * R_REP, 256, 0, stream>>>(pos, cell, bins, counts);
    rdf_finalize_kernel<<<R_REP, 128, 0, stream>>>(cell, gt, bins, counts, (float*)d_out);
}